// MessageBlock_20289425506992
// MI455X (gfx1250) — compile-verified
//
#include <hip/hip_runtime.h>

typedef float v2f __attribute__((ext_vector_type(2)));
typedef float v8f __attribute__((ext_vector_type(8)));

#define EMB   128
#define OUT3  384
#define NRBF  20
#define RCUT  5.0f
#define LDSP  (EMB + 1)   // padded LDS row to avoid bank conflicts

__device__ __forceinline__ float fast_silu(float x) {
  // x * sigmoid(x); v_rcp_f32 (1 ulp) instead of IEEE divide chain
  return x * __builtin_amdgcn_rcpf(1.0f + __expf(-x));
}

// ---------------------------------------------------------------- init d_out
// b128 streaming copy of s and v into the output (atomics accumulate on top)
__global__ __launch_bounds__(256) void init_out_kernel(
    const float4* __restrict__ s4, const float4* __restrict__ v4,
    float4* __restrict__ out4, int n_s4, int n_v4) {
  int i = blockIdx.x * blockDim.x + threadIdx.x;
  int stride = gridDim.x * blockDim.x;
  for (int k = i; k < n_s4; k += stride) out4[k] = s4[k];
  for (int k = i; k < n_v4; k += stride) out4[n_s4 + k] = v4[k];
}

// ------------------------------------------- node MLP: s_pass via f32 WMMA
// block = 256 threads = 8 waves; one block computes a 16-row tile.
__global__ __launch_bounds__(256) void node_mlp_kernel(
    const float* __restrict__ s,  const float* __restrict__ W1,
    const float* __restrict__ b1, const float* __restrict__ W2,
    const float* __restrict__ b2, float* __restrict__ s_pass, int n_nodes) {
  __shared__ float hbuf[16 * LDSP];

  const int lane  = threadIdx.x & 31;
  const int wave  = threadIdx.x >> 5;
  const int l     = lane & 15;          // M (A/C) or N (B/C) sub-index
  const int hi    = (lane >> 4) & 1;    // lane half selects K/M offset
  const int khalf = hi * 2;
  const int row0  = blockIdx.x * 16;

  // ---- stage 1: h = silu(s @ W1 + b1); wave w owns output cols [16w, 16w+16)
  {
    const int n0 = wave * 16;
    int arow = row0 + l; if (arow >= n_nodes) arow = n_nodes - 1;
    const float* srow = s + (size_t)arow * EMB;

    v8f acc;
    float bias = b1[n0 + l];
#pragma unroll
    for (int r = 0; r < 8; ++r) acc[r] = bias;

#pragma unroll 4
    for (int k = 0; k < EMB; k += 4) {
      const int kk = k + khalf;
      v2f A, B;
      // A 16x4: VGPR0 = K {0 | 2}, VGPR1 = K {1 | 3} per lane-half
      A.x = srow[kk];
      A.y = srow[kk + 1];
      // B 4x16: same K striping, N across lanes 0-15
      B.x = W1[(size_t)kk * EMB + n0 + l];
      B.y = W1[(size_t)(kk + 1) * EMB + n0 + l];
      acc = __builtin_amdgcn_wmma_f32_16x16x4_f32(
          false, A, false, B, (short)0, acc, false, false);
    }
#pragma unroll
    for (int r = 0; r < 8; ++r) {
      const int m = r + hi * 8;           // C/D layout: lanes 16-31 hold M+8
      hbuf[m * LDSP + n0 + l] = fast_silu(acc[r]);
    }
  }
  __syncthreads();

  // ---- stage 2: s_pass = h @ W2 + b2 (16x384); wave handles 3 n-tiles
#pragma unroll
  for (int j = 0; j < 3; ++j) {
    const int n0 = (wave + 8 * j) * 16;
    v8f acc;
    float bias = b2[n0 + l];
#pragma unroll
    for (int r = 0; r < 8; ++r) acc[r] = bias;

#pragma unroll 4
    for (int k = 0; k < EMB; k += 4) {
      const int kk = k + khalf;
      v2f A, B;
      A.x = hbuf[l * LDSP + kk];
      A.y = hbuf[l * LDSP + kk + 1];
      B.x = W2[(size_t)kk * OUT3 + n0 + l];
      B.y = W2[(size_t)(kk + 1) * OUT3 + n0 + l];
      acc = __builtin_amdgcn_wmma_f32_16x16x4_f32(
          false, A, false, B, (short)0, acc, false, false);
    }
#pragma unroll
    for (int r = 0; r < 8; ++r) {
      const int m    = r + hi * 8;
      const int grow = row0 + m;
      if (grow < n_nodes) s_pass[(size_t)grow * OUT3 + n0 + l] = acc[r];
    }
  }
}

// -------------------------------------------------- edge gather/scatter pass
// 128 threads per edge (thread = channel), 2 edges per 256-thread block.
__global__ __launch_bounds__(256) void edge_kernel(
    const long long* __restrict__ edges, const float* __restrict__ r_ij,
    const float* __restrict__ r_norm,    const float* __restrict__ Wr,
    const float* __restrict__ br,        const float* __restrict__ s_pass,
    const float* __restrict__ v,         float* __restrict__ out_s,
    float* __restrict__ out_v, int n_edges) {
  const int g = threadIdx.x >> 7;       // edge slot in block
  const int c = threadIdx.x & 127;      // channel
  const int e = blockIdx.x * 2 + g;

  __shared__ float rbf_sh[2][NRBF];
  __shared__ float rn_sh[2][3];
  __shared__ float fcut_sh[2];

  const bool valid = (e < n_edges);
  long long src = 0, dst = 0;
  float r = 1.0f;
  if (valid) {
    src = edges[(size_t)e * 2 + 0];
    dst = edges[(size_t)e * 2 + 1];
    r   = r_ij[e];
  }
  const float PI = 3.14159265358979323846f;
  if (valid && c < NRBF) rbf_sh[g][c] = sinf((float)(c + 1) * (PI / RCUT) * r) / r;
  if (valid && c < 3)    rn_sh[g][c]  = r_norm[(size_t)e * 3 + c];
  if (valid && c == 0)   // per-edge scalar: compute cosine cutoff once
    fcut_sh[g] = (r <= RCUT) ? 0.5f * (cosf(PI * r / RCUT) + 1.0f) : 0.0f;
  __syncthreads();
  if (!valid) return;

  const float f_cut = fcut_sh[g];

  // rbf @ Wr + br for this channel's three output slots (gate, ds, rep)
  float p0 = br[c], p1 = br[c + EMB], p2 = br[c + 2 * EMB];
#pragma unroll
  for (int n = 0; n < NRBF; ++n) {
    const float rb = rbf_sh[g][n];
    p0 = fmaf(rb, Wr[n * OUT3 + c],           p0);
    p1 = fmaf(rb, Wr[n * OUT3 + c + EMB],     p1);
    p2 = fmaf(rb, Wr[n * OUT3 + c + 2 * EMB], p2);
  }

  const size_t db = (size_t)dst * OUT3;
  const float dvg  = p0 * f_cut * s_pass[db + c];            // delta_v gate
  const float ds   = p1 * f_cut * s_pass[db + c + EMB];      // delta_s
  const float drep = p2 * f_cut * s_pass[db + c + 2 * EMB];  // delta_rep

  unsafeAtomicAdd(&out_s[(size_t)src * EMB + c], ds);
#pragma unroll
  for (int d = 0; d < 3; ++d) {
    const float dv = v[db + d * EMB + c] * dvg + rn_sh[g][d] * drep;
    unsafeAtomicAdd(&out_v[(size_t)src * OUT3 + d * EMB + c], dv);
  }
}

// ---------------------------------------------------------------------------
extern "C" void kernel_launch(void* const* d_in, const int* in_sizes, int n_in,
                              void* d_out, int out_size, void* d_ws, size_t ws_size,
                              hipStream_t stream) {
  const float*     s      = (const float*)d_in[0];
  const float*     v      = (const float*)d_in[1];
  const long long* edges  = (const long long*)d_in[2];
  const float*     r_ij   = (const float*)d_in[3];
  const float*     r_norm = (const float*)d_in[4];
  const float*     W1     = (const float*)d_in[5];
  const float*     b1     = (const float*)d_in[6];
  const float*     W2     = (const float*)d_in[7];
  const float*     b2     = (const float*)d_in[8];
  const float*     Wr     = (const float*)d_in[9];
  const float*     br     = (const float*)d_in[10];

  const int n_nodes = in_sizes[0] / EMB;
  const int n_edges = in_sizes[2] / 2;
  const int n_s = n_nodes * EMB;       // multiple of 4
  const int n_v = n_nodes * 3 * EMB;   // multiple of 4

  float* out    = (float*)d_out;
  float* out_s  = out;
  float* out_v  = out + (size_t)n_s;
  float* s_pass = (float*)d_ws;        // n_nodes * 384 floats scratch

  init_out_kernel<<<2048, 256, 0, stream>>>(
      (const float4*)s, (const float4*)v, (float4*)out, n_s / 4, n_v / 4);
  node_mlp_kernel<<<(n_nodes + 15) / 16, 256, 0, stream>>>(
      s, W1, b1, W2, b2, s_pass, n_nodes);
  edge_kernel<<<(n_edges + 1) / 2, 256, 0, stream>>>(
      edges, r_ij, r_norm, Wr, br, s_pass, v, out_s, out_v, n_edges);
}